// CodeBook_28930899706226
// MI455X (gfx1250) — compile-verified
//
#include <hip/hip_runtime.h>

typedef __attribute__((ext_vector_type(2))) float v2f;
typedef __attribute__((ext_vector_type(8))) float v8f;

#define CH        256       // code / channel dimension
#define CODES     1024      // number of codebook entries
#define HW        1024      // h*w = 32*32
#define NPIX      32768     // 32 * 32 * 32 pixels
#define PIXB      64        // pixels per block (4 waves x 16)
#define THREADS   128       // 4 wave32s
#define NTILES    (CODES / 16)
#define WS_STRIDE 260       // padded LDS row stride (floats), breaks bank conflicts

// ---------------------------------------------------------------------------
// Kernel 1: wn[j] = ||W_j||^2   (1024 rows of 256)
// ---------------------------------------------------------------------------
__global__ void wnorm_kernel(const float* __restrict__ W, float* __restrict__ wn) {
    int j = blockIdx.x * blockDim.x + threadIdx.x;
    if (j < CODES) {
        const float4* row = (const float4*)(W + (size_t)j * CH);
        float s = 0.f;
        #pragma unroll
        for (int i = 0; i < CH / 4; ++i) {
            float4 v = row[i];
            s += v.x * v.x + v.y * v.y + v.z * v.z + v.w * v.w;
        }
        wn[j] = s;
    }
}

// ---------------------------------------------------------------------------
// Kernel 2: fused GEMM (WMMA f32 16x16x4) + argmin + gather
// ---------------------------------------------------------------------------
__global__ void __launch_bounds__(THREADS)
vq_kernel(const float* __restrict__ z, const float* __restrict__ W,
          const float* __restrict__ wn,
          float* __restrict__ out_q, float* __restrict__ out_st,
          float* __restrict__ out_idx) {
    __shared__ __align__(16) float ws[2][16 * WS_STRIDE];  // double-buffered B tile
    __shared__ int idxS[PIXB];

    const int t    = threadIdx.x;
    const int lane = t & 31;
    const int wv   = t >> 5;      // wave 0..3 -> M-tile
    const int l15  = lane & 15;
    const int hi   = lane >> 4;   // 0: K pair {4k,4k+1}, 1: {4k+2,4k+3}

    const int p0   = blockIdx.x * PIXB;   // first pixel of block
    const int n    = p0 >> 10;            // image index (PIXB divides 1024)
    const int rem0 = p0 & 1023;           // h*32+w of first pixel

    // ---- Load A fragments (this wave's 16 pixels x 256 channels) into VGPRs.
    // z element (pixel p, channel c) lives at n*CH*HW + c*HW + (p & 1023).
    // A layout (V_WMMA_F32_16X16X4_F32): lanes 0-15 M=lane K={4k,4k+1};
    //                                    lanes 16-31 M=lane-16 K={4k+2,4k+3}.
    const float* zb = z + (size_t)n * (CH * HW)
                        + (size_t)(2 * hi) * HW
                        + (rem0 + wv * 16 + l15);
    v2f A[64];
    #pragma unroll
    for (int kk = 0; kk < 64; ++kk) {
        A[kk].x = zb[(size_t)(4 * kk)     * HW];
        A[kk].y = zb[(size_t)(4 * kk + 1) * HW];
    }

    // Stage one 16-code tile of W (row-major, padded) into LDS buffer `buf`.
    auto stage = [&](int jt, int buf) {
        const float4* wsrc = (const float4*)(W + (size_t)jt * 16 * CH);
        #pragma unroll
        for (int i = 0; i < (16 * CH / 4) / THREADS; ++i) {  // 8 iters
            int q   = i * THREADS + t;   // 0..1023
            int row = q >> 6;            // code row in tile
            int c4  = q & 63;            // float4 column
            float4 v = wsrc[(size_t)row * (CH / 4) + c4];
            *(float4*)&ws[buf][row * WS_STRIDE + c4 * 4] = v;
        }
    };

    float best[8];
    int   bidx[8];
    #pragma unroll
    for (int r = 0; r < 8; ++r) { best[r] = 3.0e38f; bidx[r] = 0; }

    stage(0, 0);

    #pragma unroll 1
    for (int jt = 0; jt < NTILES; ++jt) {
        const int buf = jt & 1;
        __syncthreads();                         // buf staged; prev reads done
        if (jt + 1 < NTILES) stage(jt + 1, buf ^ 1);

        // B layout mirrors A: lanes 0-15 N=lane K={4k,4k+1}; lanes 16-31 K={4k+2,4k+3}.
        const float* wrow = &ws[buf][l15 * WS_STRIDE + 2 * hi];

        v8f acc0 = {};
        v8f acc1 = {};
        #pragma unroll
        for (int kk = 0; kk < 64; kk += 2) {
            v2f b0 = *(const v2f*)(wrow + 4 * kk);
            acc0 = __builtin_amdgcn_wmma_f32_16x16x4_f32(
                false, A[kk], false, b0, (short)0, acc0, false, false);
            v2f b1 = *(const v2f*)(wrow + 4 * (kk + 1));
            acc1 = __builtin_amdgcn_wmma_f32_16x16x4_f32(
                false, A[kk + 1], false, b1, (short)0, acc1, false, false);
        }

        // score = ||w_j||^2 - 2 * (z . w_j); ||z||^2 is row-constant -> dropped
        const int   j  = jt * 16 + l15;
        const float wj = wn[j];
        #pragma unroll
        for (int r = 0; r < 8; ++r) {
            float s = wj - 2.0f * (acc0[r] + acc1[r]);
            if (s < best[r]) { best[r] = s; bidx[r] = j; }   // strict '<' keeps first index
        }
    }

    // ---- Cross-lane argmin over the 16 lanes sharing each output row.
    // Lanes 0-15 hold rows 0-7; lanes 16-31 hold rows 8-15 (C/D layout).
    #pragma unroll
    for (int m = 1; m <= 8; m <<= 1) {
        #pragma unroll
        for (int r = 0; r < 8; ++r) {
            float ov = __shfl_xor(best[r], m, 32);
            int   oi = __shfl_xor(bidx[r], m, 32);
            if (ov < best[r] || (ov == best[r] && oi < bidx[r])) {
                best[r] = ov; bidx[r] = oi;
            }
        }
    }

    if (l15 == 0) {
        #pragma unroll
        for (int r = 0; r < 8; ++r) {
            int row = r + 8 * hi;
            int pix = wv * 16 + row;
            idxS[pix] = bidx[r];
            out_idx[p0 + pix] = (float)bidx[r];
        }
    }
    __syncthreads();

    // ---- Gather: quantized == straight_through == W[idx], back to (n,c,h,w).
    #pragma unroll 1
    for (int i = 0; i < (PIXB * CH) / THREADS; ++i) {  // 128 iters
        int q = i * THREADS + t;
        int c = q >> 6;           // PIXB == 64
        int m = q & 63;
        float v = W[(size_t)idxS[m] * CH + c];
        size_t off = (size_t)n * (CH * HW) + (size_t)c * HW + (rem0 + m);
        out_q[off]  = v;
        out_st[off] = v;
    }
}

// ---------------------------------------------------------------------------
extern "C" void kernel_launch(void* const* d_in, const int* in_sizes, int n_in,
                              void* d_out, int out_size, void* d_ws, size_t ws_size,
                              hipStream_t stream) {
    const float* z = (const float*)d_in[0];        // (32, 256, 32, 32)
    const float* W = (const float*)d_in[1];        // (1024, 256)

    float* wn      = (float*)d_ws;                 // 1024 floats scratch
    float* out     = (float*)d_out;
    float* out_q   = out;                          // quantized      (8388608)
    float* out_st  = out + (size_t)NPIX * CH;      // straight_thru  (8388608)
    float* out_idx = out + (size_t)2 * NPIX * CH;  // indices        (32768)

    wnorm_kernel<<<CODES / 128, 128, 0, stream>>>(W, wn);
    vq_kernel<<<NPIX / PIXB, THREADS, 0, stream>>>(z, W, wn, out_q, out_st, out_idx);
}